// ALiBi_46462956208773
// MI455X (gfx1250) — compile-verified
//
#include <hip/hip_runtime.h>
#include <cmath>

// ALiBi bias materialization for MI455X (gfx1250, wave32).
//
// out[h, i, j] = slope[h] * (j - i)   if j <= i
//              = -inf                  if j >  i
//
// Store-bandwidth bound: 1 GiB of fp32 stores -> ~46us floor at 23.3 TB/s.
// Each wave32 produces one 16x16 tile with a single rank-1
// V_WMMA_F32_16X16X4_F32 (row term folded into the C accumulator), applies
// the causal mask with cndmask selects, and streams the tile out with
// non-temporal b32 stores (output >> 192 MB L2, so bypass caching).

typedef __attribute__((ext_vector_type(2))) float v2f;
typedef __attribute__((ext_vector_type(8))) float v8f;

__global__ __launch_bounds__(256) void alibi_wmma_kernel(
    const float* __restrict__ slopes,
    float* __restrict__ out,
    int L,              // sequence length (4096)
    int tilesPerRow,    // L / 16
    int tilesPerHead,   // (L/16)^2
    int totalTiles)     // H * (L/16)^2
{
    const int lane = threadIdx.x & 31;
    const int wave = threadIdx.x >> 5;
    const int tile = blockIdx.x * 8 + wave;   // one 16x16 tile per wave
    if (tile >= totalTiles) return;           // wave-uniform guard

    const int h  = tile / tilesPerHead;
    const int r  = tile - h * tilesPerHead;
    const int ti = r / tilesPerRow;
    const int tj = r - ti * tilesPerRow;
    const int i0 = ti << 4;                   // tile row base
    const int j0 = tj << 4;                   // tile col base

    const float slope = slopes[h];

    // C/D layout (32-bit 16x16): lane<16 -> M = vgpr_r,   N = lane
    //                            lane>=16 -> M = vgpr_r+8, N = lane-16
    const int  n  = lane & 15;
    const bool hi = lane >= 16;
    const int  j  = j0 + n;                   // this lane's global column
    const int  mb = i0 + (hi ? 8 : 0);        // row base for this lane's 8 regs

    // Rank-1 WMMA: D = A(16x4) x B(4x16) + C, only the K=0 row is nonzero.
    // A 16x4 layout: lanes 0-15 hold {K=0,K=1}, lanes 16-31 hold {K=2,K=3};
    // all K>=1 coefficients are zero, so B's K>=1 lane placement is moot.
    v2f a; a.x = hi ? 0.0f : 1.0f;               a.y = 0.0f;
    v2f b; b.x = hi ? 0.0f : slope * (float)j;   b.y = 0.0f;
    v8f c;
#pragma unroll
    for (int rr = 0; rr < 8; ++rr)
        c[rr] = -slope * (float)(mb + rr);       // row term in accumulator

    // D[M][N] = slope*(j0+N) - slope*(i0+M)
    v8f d = __builtin_amdgcn_wmma_f32_16x16x4_f32(
        /*neg_a=*/false, a, /*neg_b=*/false, b,
        /*c_mod=*/(short)0, c, /*reuse_a=*/false, /*reuse_b=*/false);

    const float NEG_INF = -__builtin_inff();
    float* outp = out + ((size_t)h * (size_t)L + (size_t)mb) * (size_t)L + j;
#pragma unroll
    for (int rr = 0; rr < 8; ++rr) {
        float v = (j > mb + rr) ? NEG_INF : d[rr];   // causal mask
        __builtin_nontemporal_store(v, outp + (size_t)rr * (size_t)L);
    }
}

extern "C" void kernel_launch(void* const* d_in, const int* in_sizes, int n_in,
                              void* d_out, int out_size, void* d_ws, size_t ws_size,
                              hipStream_t stream) {
    (void)n_in; (void)d_ws; (void)ws_size;

    const float* slopes = (const float*)d_in[0];  // [H] fp32 slopes
    const int H = in_sizes[0];                    // 16

    // out_size = H * L * L  ->  recover L host-side (seq_len lives on device).
    const long long LL = (long long)out_size / (long long)(H > 0 ? H : 1);
    const int L = (int)llround(sqrt((double)LL)); // 4096

    const int tilesPerRow  = L / 16;                          // 256
    const int tilesPerHead = tilesPerRow * tilesPerRow;       // 65536
    const long long totalTiles = (long long)tilesPerHead * H; // 1,048,576
    const long long blocks = (totalTiles + 7) / 8;            // 8 waves/block

    alibi_wmma_kernel<<<dim3((unsigned)blocks), 256, 0, stream>>>(
        slopes, (float*)d_out, L, tilesPerRow, tilesPerHead, (int)totalTiles);
}